// CrystalDecoder_27522150432770
// MI455X (gfx1250) — compile-verified
//
#include <hip/hip_runtime.h>

// CrystalDecoder for MI455X (gfx1250), wave32 + WMMA f32 16x16x4.
//
// Factorization: z_expanded has only B=256 distinct rows, so
//  - edge MLP output depends only on (graph_id[src], graph_id[dst]) -> 256x256x4 LUT
//  - z-contribution to node decoder folds into ZN = z_proj @ nd1_w  (256x64)
// Remaining heavy work: per-node GEMMs (N x 128 x 128) and (N x 128 x 64) done with
// v_wmma_f32_16x16x4_f32; weights pre-swizzled into LDS B-fragment order; each wave
// carries TWO 16-node tiles so every B fragment feeds 2 WMMAs. The node kernel is
// persistent (grid-stride) so the 96KB weight staging happens once per block, and
// the next iteration's node_emb rows are prefetched (global_prefetch_b8) while the
// current WMMA phase runs.

typedef __attribute__((ext_vector_type(2))) float v2f;
typedef __attribute__((ext_vector_type(8))) float v8f;

#define LATENT 32
#define HID    64
#define TWOH   128
#define NPW    16      // nodes per tile (WMMA M)
#define TPW    2       // tiles per wave
#define WAVES  8       // waves per block
#define NPB    (WAVES * TPW * NPW)   // 256 nodes per block-iteration
#define WSTR   132     // LDS row stride (words); 132 % 64 == 4 -> conflict-free frags

// ---------------------------------------------------------------------------
// Kernel S: per-graph small MLPs. One block per graph b (128 threads).
// ---------------------------------------------------------------------------
__global__ __launch_bounds__(128) void k_graph_small(
    const float* __restrict__ z,
    const float* __restrict__ lp_w, const float* __restrict__ lp_b,
    const float* __restrict__ ed1_w, const float* __restrict__ nd1_w,
    const float* __restrict__ en1_w, const float* __restrict__ en1_b,
    const float* __restrict__ en2_w, const float* __restrict__ en2_b,
    const float* __restrict__ st1_w, const float* __restrict__ st1_b,
    const float* __restrict__ st2_w, const float* __restrict__ st2_b,
    float* __restrict__ Atab, float* __restrict__ Btab, float* __restrict__ ZN,
    float* __restrict__ pe, float* __restrict__ ps)
{
    __shared__ float zsh[LATENT];
    __shared__ float zpsh[TWOH];
    __shared__ float h1[HID];
    __shared__ float h2[HID];
    const int b = blockIdx.x, tid = threadIdx.x;

    if (tid < LATENT) zsh[tid] = z[b * LATENT + tid];
    __syncthreads();

    // z_proj = relu(z @ lp_w + lp_b)
    {
        float s = lp_b[tid];
        for (int k = 0; k < LATENT; ++k) s += zsh[k] * lp_w[k * TWOH + tid];
        zpsh[tid] = fmaxf(s, 0.f);
    }
    // energy / stress hidden layers
    {
        const int c = tid & (HID - 1);
        if (tid < HID) {
            float s = en1_b[c];
            for (int k = 0; k < LATENT; ++k) s += zsh[k] * en1_w[k * HID + c];
            h1[c] = fmaxf(s, 0.f);
        } else {
            float s = st1_b[c];
            for (int k = 0; k < LATENT; ++k) s += zsh[k] * st1_w[k * HID + c];
            h2[c] = fmaxf(s, 0.f);
        }
    }
    __syncthreads();

    // edge tables: A = zp @ ed1_w[0:128,:], B = zp @ ed1_w[128:256,:]
    {
        const int c = tid & (HID - 1);
        const float* wcol = (tid < HID) ? (ed1_w + c) : (ed1_w + TWOH * HID + c);
        float s = 0.f;
        for (int k = 0; k < TWOH; ++k) s += zpsh[k] * wcol[k * HID];
        if (tid < HID) Atab[b * HID + c] = s; else Btab[b * HID + c] = s;
    }
    // ZN = zp @ nd1_w  (z contribution to node decoder L1)
    if (tid < HID) {
        float s = 0.f;
        for (int k = 0; k < TWOH; ++k) s += zpsh[k] * nd1_w[k * HID + tid];
        ZN[b * HID + tid] = s;
    }
    // heads
    if (tid < 2) {
        float s = en2_b[tid];
        for (int k = 0; k < HID; ++k) s += h1[k] * en2_w[k * 2 + tid];
        pe[b * 2 + tid] = s;
    } else if (tid >= 64 && tid < 73) {
        const int o = tid - 64;
        float s = st2_b[o];
        for (int k = 0; k < HID; ++k) s += h2[k] * st2_w[k * 9 + o];
        ps[b * 9 + o] = s;
    }
}

// ---------------------------------------------------------------------------
// Kernel P: pair table T4[gs][gd] = edge MLP output for graph-id pair (gs,gd).
// ---------------------------------------------------------------------------
__global__ __launch_bounds__(256) void k_pair_table(
    const float* __restrict__ Atab, const float* __restrict__ Btab,
    const float* __restrict__ ed1_b,
    const float* __restrict__ ed2_w, const float* __restrict__ ed2_b,
    float* __restrict__ T4, int Bg)
{
    __shared__ float Ash[HID];
    __shared__ float wsh[HID * 3];
    const int gs = blockIdx.x, gd = threadIdx.x;

    if (gd < HID) Ash[gd] = Atab[gs * HID + gd] + ed1_b[gd];
    if (gd >= 64 && gd < 64 + HID * 3) wsh[gd - 64] = ed2_w[gd - 64];
    __syncthreads();

    const float* Brow = Btab + (size_t)gd * HID;
    float o0 = ed2_b[0], o1 = ed2_b[1], o2 = ed2_b[2];
    for (int k = 0; k < HID; ++k) {
        const float h = fmaxf(Ash[k] + Brow[k], 0.f);
        o0 += h * wsh[k * 3 + 0];
        o1 += h * wsh[k * 3 + 1];
        o2 += h * wsh[k * 3 + 2];
    }
    ((float4*)T4)[(size_t)gs * Bg + gd] = make_float4(o0, o1, o2, 0.f);
}

// ---------------------------------------------------------------------------
// Kernel N: node decoder. Persistent; 256 threads = 8 waves; 2 x 16-node tiles/wave.
// ---------------------------------------------------------------------------
__global__ __launch_bounds__(256) void k_nodes(
    const float* __restrict__ node_emb, const int* __restrict__ graph_id,
    const float* __restrict__ nep_w, const float* __restrict__ nep_b,
    const float* __restrict__ nd1_w, const float* __restrict__ nd1_b,
    const float* __restrict__ nd2_w, const float* __restrict__ nd2_b,
    const float* __restrict__ ZN,
    float* __restrict__ out, int N)
{
    extern __shared__ float smem[];
    float* ws1      = smem;                    // 16384: nep_w in B-fragment order
    float* ws2      = ws1 + 16384;             //  8192: nd1_w in B-fragment order
    float* w3sh     = ws2 + 8192;              //   256: nd2_w [64][4]
    float* nb_sh    = w3sh + 256;              //   128: nep_b
    float* d1b_sh   = nb_sh + 128;             //    64: nd1_b
    float* d2b_sh   = d1b_sh + 64;             //     4: nd2_b
    int*   gid_sh   = (int*)(d2b_sh + 4);      //   256: per-wave graph ids (32/wave)
    float* wbuf_all = (float*)(gid_sh + 256);  // 8*2*16*132 per-wave tile buffers

    const int tid = threadIdx.x;

    // ---- stage weights into LDS once per block, pre-swizzled B-fragment order ----
    // B frag (kb,j): lane L needs W[kb*4 + 2*(L>>4) + t][16*j + (L&15)], t in {0,1}
    for (int idx = tid; idx < TWOH * TWOH; idx += 256) {
        const int k = idx >> 7, n = idx & 127;
        const int kb = k >> 2, rr = k & 3;
        const int L = ((rr >> 1) << 4) | (n & 15);
        ws1[(((kb << 3) + (n >> 4)) * 32 + L) * 2 + (rr & 1)] = nep_w[idx];
    }
    for (int idx = tid; idx < TWOH * HID; idx += 256) {
        const int k = idx >> 6, n = idx & 63;
        const int kb = k >> 2, rr = k & 3;
        const int L = ((rr >> 1) << 4) | (n & 15);
        ws2[(((kb << 2) + (n >> 4)) * 32 + L) * 2 + (rr & 1)] = nd1_w[idx];
    }
    w3sh[tid] = nd2_w[tid];
    if (tid < 128) nb_sh[tid] = nep_b[tid];
    if (tid < 64)  d1b_sh[tid] = nd1_b[tid];
    if (tid < 4)   d2b_sh[tid] = nd2_b[tid];
    __syncthreads();

    const int wv   = tid >> 5;
    const int lane = tid & 31;
    const int row  = lane & 15;                // A row / C-D col / out node
    const int half = lane >> 4;
    const int koff = half * 2;

    float* wb0 = wbuf_all + wv * (TPW * NPW * WSTR);
    float* wb1 = wb0 + NPW * WSTR;

    const v8f z8 = {0.f, 0.f, 0.f, 0.f, 0.f, 0.f, 0.f, 0.f};
    const int stride = (int)gridDim.x * NPB;

    // ---- persistent grid-stride loop over 256-node chunks ----
    for (int base = (int)blockIdx.x * NPB; base < N; base += stride) {
        const int nb0 = base + wv * (TPW * NPW);
        if (nb0 >= N) continue;                // wave-uniform
        const int nb1 = nb0 + NPW;
        const bool two = (nb1 < N);            // N is a multiple of 16

        if (lane < 16) {
            gid_sh[wv * 32 + lane]      = graph_id[nb0 + lane];
            gid_sh[wv * 32 + 16 + lane] = two ? graph_id[nb1 + lane] : 0;
        }

        // stage node_emb tiles [16 x 128] into LDS (coalesced b128)
        for (int r = 0; r < 16; ++r)
            *(float4*)(wb0 + r * WSTR + lane * 4) =
                ((const float4*)(node_emb + (size_t)(nb0 + r) * TWOH))[lane];
        if (two)
            for (int r = 0; r < 16; ++r)
                *(float4*)(wb1 + r * WSTR + lane * 4) =
                    ((const float4*)(node_emb + (size_t)(nb1 + r) * TWOH))[lane];

        // prefetch next iteration's 32 rows (16KB) while WMMAs run
        {
            const int nxt = nb0 + stride;
            if (nxt + TPW * NPW <= N) {
                const char* pb = (const char*)(node_emb + (size_t)nxt * TWOH);
#pragma unroll
                for (int k = 0; k < 4; ++k)
                    __builtin_prefetch(pb + (size_t)(lane + k * 32) * 128, 0, 0);
            }
        }

        // ---- stage 1: nep = relu(node_emb @ nep_w + nep_b)  (2 x 16x128, K=128) ----
        v8f acc0[8], acc1[8];
#pragma unroll
        for (int j = 0; j < 8; ++j) { acc0[j] = z8; acc1[j] = z8; }
#pragma unroll 4
        for (int kb = 0; kb < 32; ++kb) {
            const v2f a0 = *(const v2f*)(wb0 + row * WSTR + kb * 4 + koff);
            const v2f a1 = *(const v2f*)(wb1 + row * WSTR + kb * 4 + koff);
            v2f b[8];
#pragma unroll
            for (int j = 0; j < 8; ++j)
                b[j] = *(const v2f*)(ws1 + (((kb << 3) + j) * 32 + lane) * 2);
#pragma unroll
            for (int j = 0; j < 8; ++j) {
                acc0[j] = __builtin_amdgcn_wmma_f32_16x16x4_f32(
                    false, a0, false, b[j], (short)0, acc0[j], false, false);
                acc1[j] = __builtin_amdgcn_wmma_f32_16x16x4_f32(
                    false, a1, false, b[j], (short)0, acc1[j], false, false);
            }
        }
        // epilogue: relu + bias, write h back into the tile buffers (DS in-order)
#pragma unroll
        for (int j = 0; j < 8; ++j) {
            const int n = j * 16 + row;
            const float bb = nb_sh[n];
#pragma unroll
            for (int r = 0; r < 8; ++r) {
                const int m = r + half * 8;
                wb0[m * WSTR + n] = fmaxf(acc0[j][r] + bb, 0.f);
                wb1[m * WSTR + n] = fmaxf(acc1[j][r] + bb, 0.f);
            }
        }

        // ---- stage 2: t = relu(h @ nd1_w + nd1_b + ZN[graph_id])  (2 x 16x64) ----
        v8f tc0[4], tc1[4];
#pragma unroll
        for (int j = 0; j < 4; ++j) { tc0[j] = z8; tc1[j] = z8; }
#pragma unroll 4
        for (int kb = 0; kb < 32; ++kb) {
            const v2f a0 = *(const v2f*)(wb0 + row * WSTR + kb * 4 + koff);
            const v2f a1 = *(const v2f*)(wb1 + row * WSTR + kb * 4 + koff);
            v2f b[4];
#pragma unroll
            for (int j = 0; j < 4; ++j)
                b[j] = *(const v2f*)(ws2 + (((kb << 2) + j) * 32 + lane) * 2);
#pragma unroll
            for (int j = 0; j < 4; ++j) {
                tc0[j] = __builtin_amdgcn_wmma_f32_16x16x4_f32(
                    false, a0, false, b[j], (short)0, tc0[j], false, false);
                tc1[j] = __builtin_amdgcn_wmma_f32_16x16x4_f32(
                    false, a1, false, b[j], (short)0, tc1[j], false, false);
            }
        }
#pragma unroll
        for (int j = 0; j < 4; ++j) {
            const int n = j * 16 + row;
            const float bb = d1b_sh[n];
#pragma unroll
            for (int r = 0; r < 8; ++r) {
                const int m  = r + half * 8;
                const int g0 = gid_sh[wv * 32 + m];
                const int g1 = gid_sh[wv * 32 + 16 + m];
                wb0[m * WSTR + n] = fmaxf(tc0[j][r] + bb + ZN[g0 * HID + n], 0.f);
                wb1[m * WSTR + n] = fmaxf(tc1[j][r] + bb + ZN[g1 * HID + n], 0.f);
            }
        }

        // ---- stage 3: out = t @ nd2_w + nd2_b  (VALU; node = row, cols 2*half..) ----
        const float b0 = d2b_sh[half * 2], b1 = d2b_sh[half * 2 + 1];
        float o00 = b0, o01 = b1, o10 = b0, o11 = b1;
        for (int k = 0; k < HID; ++k) {
            const float w0 = w3sh[k * 4 + half * 2];
            const float w1 = w3sh[k * 4 + half * 2 + 1];
            const float t0 = wb0[row * WSTR + k];
            const float t1 = wb1[row * WSTR + k];
            o00 += t0 * w0; o01 += t0 * w1;
            o10 += t1 * w0; o11 += t1 * w1;
        }
        *(float2*)(out + (size_t)(nb0 + row) * 4 + half * 2) = make_float2(o00, o01);
        if (two)
            *(float2*)(out + (size_t)(nb1 + row) * 4 + half * 2) = make_float2(o10, o11);
    }
}

// ---------------------------------------------------------------------------
// Kernel E: edges = pure gather through the pair LUT. 4 edges/thread.
// ---------------------------------------------------------------------------
__global__ __launch_bounds__(256) void k_edges(
    const int* __restrict__ src, const int* __restrict__ dst,
    const int* __restrict__ gid, const float* __restrict__ T4,
    float* __restrict__ out, int E, int Bg)
{
    const int i = blockIdx.x * blockDim.x + threadIdx.x;
    const int e0 = i * 4;
    if (e0 + 3 < E) {
        const int4 s4 = ((const int4*)src)[i];
        const int4 d4 = ((const int4*)dst)[i];
        const float4* T = (const float4*)T4;
        const float4 t0 = T[(size_t)gid[s4.x] * Bg + gid[d4.x]];
        const float4 t1 = T[(size_t)gid[s4.y] * Bg + gid[d4.y]];
        const float4 t2 = T[(size_t)gid[s4.z] * Bg + gid[d4.z]];
        const float4 t3 = T[(size_t)gid[s4.w] * Bg + gid[d4.w]];
        float4* o = (float4*)(out + (size_t)e0 * 3);
        o[0] = make_float4(t0.x, t0.y, t0.z, t1.x);
        o[1] = make_float4(t1.y, t1.z, t2.x, t2.y);
        o[2] = make_float4(t2.z, t3.x, t3.y, t3.z);
    } else if (e0 < E) {
        for (int e = e0; e < E; ++e) {
            const float4 t = ((const float4*)T4)[(size_t)gid[src[e]] * Bg + gid[dst[e]]];
            out[(size_t)e * 3 + 0] = t.x;
            out[(size_t)e * 3 + 1] = t.y;
            out[(size_t)e * 3 + 2] = t.z;
        }
    }
}

// ---------------------------------------------------------------------------
extern "C" void kernel_launch(void* const* d_in, const int* in_sizes, int n_in,
                              void* d_out, int out_size, void* d_ws, size_t ws_size,
                              hipStream_t stream)
{
    const float* z        = (const float*)d_in[0];
    const float* node_emb = (const float*)d_in[1];
    const int*   graph_id = (const int*)d_in[2];
    const int*   src      = (const int*)d_in[3];
    const int*   dst      = (const int*)d_in[4];
    const float* lp_w  = (const float*)d_in[5];   const float* lp_b  = (const float*)d_in[6];
    const float* nep_w = (const float*)d_in[7];   const float* nep_b = (const float*)d_in[8];
    const float* nd1_w = (const float*)d_in[9];   const float* nd1_b = (const float*)d_in[10];
    const float* nd2_w = (const float*)d_in[11];  const float* nd2_b = (const float*)d_in[12];
    const float* ed1_w = (const float*)d_in[13];  const float* ed1_b = (const float*)d_in[14];
    const float* ed2_w = (const float*)d_in[15];  const float* ed2_b = (const float*)d_in[16];
    const float* en1_w = (const float*)d_in[17];  const float* en1_b = (const float*)d_in[18];
    const float* en2_w = (const float*)d_in[19];  const float* en2_b = (const float*)d_in[20];
    const float* st1_w = (const float*)d_in[21];  const float* st1_b = (const float*)d_in[22];
    const float* st2_w = (const float*)d_in[23];  const float* st2_b = (const float*)d_in[24];

    const int Bg = in_sizes[0] / LATENT;  // 256 graphs
    const int N  = in_sizes[2];           // 200000 nodes
    const int E  = in_sizes[3];           // 1500000 edges

    // workspace layout (floats)
    float* ws   = (float*)d_ws;
    float* Atab = ws;                          ws += (size_t)Bg * HID;
    float* Btab = ws;                          ws += (size_t)Bg * HID;
    float* ZN   = ws;                          ws += (size_t)Bg * HID;
    float* T4   = ws;                          // Bg*Bg*4 floats (1 MB, L2-resident)

    // output layout: recon_node [N,4] | recon_edge [E,3] | energy [Bg,2] | stress [Bg,9]
    float* out_node = (float*)d_out;
    float* out_edge = out_node + (size_t)N * 4;
    float* out_pe   = out_edge + (size_t)E * 3;
    float* out_ps   = out_pe   + (size_t)Bg * 2;

    k_graph_small<<<Bg, 128, 0, stream>>>(z, lp_w, lp_b, ed1_w, nd1_w,
                                          en1_w, en1_b, en2_w, en2_b,
                                          st1_w, st1_b, st2_w, st2_b,
                                          Atab, Btab, ZN, out_pe, out_ps);

    k_pair_table<<<Bg, Bg, 0, stream>>>(Atab, Btab, ed1_b, ed2_w, ed2_b, T4, Bg);

    {
        int blocks = (N + NPB - 1) / NPB;
        if (blocks > 512) blocks = 512;        // persistent: amortize weight staging
        const size_t lds_words = 16384 + 8192 + 256 + 128 + 64 + 4 + 256
                               + (size_t)WAVES * TPW * NPW * WSTR;
        k_nodes<<<blocks, 256, lds_words * sizeof(float), stream>>>(
            node_emb, graph_id, nep_w, nep_b, nd1_w, nd1_b, nd2_w, nd2_b,
            ZN, out_node, N);
    }
    {
        const int nthreads = (E + 3) / 4;
        const int blocks = (nthreads + 255) / 256;
        k_edges<<<blocks, 256, 0, stream>>>(src, dst, graph_id, T4, out_edge, E, Bg);
    }
}